// LightGCN_9380208574849
// MI455X (gfx1250) — compile-verified
//
#include <hip/hip_runtime.h>
#include <stdint.h>

#define DEMB   64
#define CHUNK  1024          // edges per TDM tile (2 rows x CHUNK int32 = 8 KB)
#define TPB    256

typedef uint32_t u32x4 __attribute__((ext_vector_type(4)));
typedef int      i32x8 __attribute__((ext_vector_type(8)));
typedef int      i32x4 __attribute__((ext_vector_type(4)));

// ---------------------------------------------------------------------------
// TDM: load a [2 x tile0] tile of the [2,E] int32 edge matrix (row stride E)
// into LDS, rows packed back-to-back.  One descriptor, tracked by TENSORcnt.
// ---------------------------------------------------------------------------
__device__ __forceinline__ void tdm_load_edge_tile(const int* gsrc, int* lds_dst,
                                                   int tile0, int E) {
  const uint32_t lds_off = (uint32_t)(uintptr_t)lds_dst;   // LDS byte offset
  const uint64_t gaddr   = (uint64_t)(uintptr_t)gsrc;
  const uint32_t td0     = (uint32_t)E;       // tensor_dim0 (elements)
  const uint64_t s0      = (uint64_t)E;       // tensor_dim0_stride (elements)
  const uint64_t s1      = (uint64_t)E * 2;   // tensor_dim1_stride (unused, 2D)

  // D# group 0: [1:0]=count=1, [63:32]=lds_addr, [120:64]=global_addr,
  //             [127:126]=type=2
  u32x4 g0 = { 1u,
               lds_off,
               (uint32_t)gaddr,
               (uint32_t)((gaddr >> 32) & 0x01FFFFFFu) | (2u << 30) };

  // D# group 1: wg_mask=0, data_size=2(4B), tensor_dim0=E, tensor_dim1=2,
  //             tile_dim0=tile0, tile_dim1=2, tile_dim2=0, dim0_stride=E
  i32x8 g1 = { (int)(2u << 16),
               (int)((td0 & 0xFFFFu) << 16),
               (int)(((td0 >> 16) & 0xFFFFu) | (2u << 16)),
               (int)(((uint32_t)tile0 & 0xFFFFu) << 16),
               2,
               (int)(uint32_t)s0,
               (int)(((uint32_t)(s0 >> 32) & 0xFFFFu) |
                     (((uint32_t)s1 & 0xFFFFu) << 16)),
               (int)(uint32_t)(s1 >> 16) };

  i32x4 gz4 = { 0, 0, 0, 0 };
  i32x8 gz8 = { 0, 0, 0, 0, 0, 0, 0, 0 };
  __builtin_amdgcn_tensor_load_to_lds(g0, g1, gz4, gz4, gz8, 0);
}

__device__ __forceinline__ void wait_tensorcnt_0() {
#if __has_builtin(__builtin_amdgcn_s_wait_tensorcnt)
  __builtin_amdgcn_s_wait_tensorcnt(0);
#else
  asm volatile("s_wait_tensorcnt 0x0" ::: "memory");
#endif
}
__device__ __forceinline__ void wait_tensorcnt_1() {
#if __has_builtin(__builtin_amdgcn_s_wait_tensorcnt)
  __builtin_amdgcn_s_wait_tensorcnt(1);
#else
  asm volatile("s_wait_tensorcnt 0x1" ::: "memory");
#endif
}

// ---------------------------------------------------------------------------
// K0: acc = x = emb ; buf = 0 ; deg = 0        (float4 streaming)
// ---------------------------------------------------------------------------
__global__ void lg_init(const float4* __restrict__ emb, float4* __restrict__ acc,
                        float4* __restrict__ x, float4* __restrict__ buf,
                        float* __restrict__ deg, int N) {
  const int i = blockIdx.x * blockDim.x + threadIdx.x;
  const int total4 = N * (DEMB / 4);
  if (i < total4) {
    const float4 v = emb[i];
    acc[i] = v;
    x[i]   = v;
    buf[i] = make_float4(0.0f, 0.0f, 0.0f, 0.0f);
  }
  if (i < N) deg[i] = 0.0f;
}

// ---------------------------------------------------------------------------
// K1: deg[row[e]] += 1        (hardware f32 atomic, no-return -> STOREcnt)
// ---------------------------------------------------------------------------
__global__ void lg_degree(const int* __restrict__ row, float* __restrict__ deg,
                          int E) {
  const int i = blockIdx.x * blockDim.x + threadIdx.x;
  if (i < E) unsafeAtomicAdd(&deg[row[i]], 1.0f);
}

// ---------------------------------------------------------------------------
// K2: dinv2[n] = (deg>0) ? (deg^-1/2)^2 : 0
// (reference applies dinv[row] twice per layer -> fold into a single dinv^2)
// ---------------------------------------------------------------------------
__global__ void lg_dinv(const float* __restrict__ deg, float* __restrict__ dinv2,
                        int N) {
  const int i = blockIdx.x * blockDim.x + threadIdx.x;
  if (i < N) {
    const float d = deg[i];
    const float s = (d > 0.0f) ? rsqrtf(d) : 0.0f;
    dinv2[i] = s * s;
  }
}

// ---------------------------------------------------------------------------
// Gather x[col] (16 lanes x float4 = one 256B coalesced segment per edge) and
// atomically accumulate into buf[row].
// ---------------------------------------------------------------------------
__device__ __forceinline__ void process_chunk(const int* __restrict__ sbuf,
                                              int count,
                                              const float* __restrict__ x,
                                              float* __restrict__ buf) {
  const int lane4 = (threadIdx.x & 15) * 4;
  for (int le = threadIdx.x >> 4; le < count; le += TPB / 16) {
    const int r = sbuf[le];
    const int c = sbuf[count + le];     // col row packed right after row row
    const float4 v = *(const float4*)(x + (size_t)c * DEMB + lane4);
    float* dst = buf + (size_t)r * DEMB + lane4;
    unsafeAtomicAdd(dst + 0, v.x);
    unsafeAtomicAdd(dst + 1, v.y);
    unsafeAtomicAdd(dst + 2, v.z);
    unsafeAtomicAdd(dst + 3, v.w);
  }
}

// ---------------------------------------------------------------------------
// K3: scatter layer, double-buffered TDM.  Each block owns 2 chunks: issue
// both DMAs, s_wait_tensorcnt<=1 (in-order completion -> tile 0 resident),
// process tile 0 while tile 1's DMA is in flight, then s_wait_tensorcnt 0.
// ---------------------------------------------------------------------------
__global__ void lg_scatter(const int* __restrict__ edges,
                           const float* __restrict__ x,
                           float* __restrict__ buf, int E) {
  __shared__ int sidx0[2 * CHUNK];
  __shared__ int sidx1[2 * CHUNK];

  const int64_t base0 = (int64_t)blockIdx.x * (2 * CHUNK);
  const int64_t base1 = base0 + CHUNK;
  const int64_t rem0  = (int64_t)E - base0;
  const int64_t rem1  = (int64_t)E - base1;
  const int count0 = (rem0 >= CHUNK) ? CHUNK : (rem0 > 0 ? (int)rem0 : 0);
  const int count1 = (rem1 >= CHUNK) ? CHUNK : (rem1 > 0 ? (int)rem1 : 0);

  if (threadIdx.x < 32) {               // wave 0 drives the TDM pipeline
    tdm_load_edge_tile(edges + base0, sidx0, count0, E);
    if (count1 > 0) {
      tdm_load_edge_tile(edges + base1, sidx1, count1, E);
      wait_tensorcnt_1();               // tile 0 done; tile 1 still in flight
    } else {
      wait_tensorcnt_0();
    }
  }
  __syncthreads();

  process_chunk(sidx0, count0, x, buf);

  if (count1 > 0) {                     // block-uniform condition
    if (threadIdx.x < 32) wait_tensorcnt_0();
    __syncthreads();
    process_chunk(sidx1, count1, x, buf);
  }
}

// ---------------------------------------------------------------------------
// K4: v = buf * dinv2[node]; x = v; acc += v; buf = 0; last layer: acc *= 1/4
// (float4 streaming; 4 consecutive elements share one node since rows are 64)
// ---------------------------------------------------------------------------
__global__ void lg_post(float4* __restrict__ acc, float4* __restrict__ x,
                        float4* __restrict__ buf, const float* __restrict__ dinv2,
                        int N, int last) {
  const int i = blockIdx.x * blockDim.x + threadIdx.x;
  if (i >= N * (DEMB / 4)) return;
  const float s = dinv2[i >> 4];
  const float4 b = buf[i];
  float4 v = make_float4(b.x * s, b.y * s, b.z * s, b.w * s);
  buf[i] = make_float4(0.0f, 0.0f, 0.0f, 0.0f);
  float4 a = acc[i];
  a.x += v.x; a.y += v.y; a.z += v.z; a.w += v.w;
  x[i] = v;
  if (last) { a.x *= 0.25f; a.y *= 0.25f; a.z *= 0.25f; a.w *= 0.25f; }
  acc[i] = a;
}

// ---------------------------------------------------------------------------
extern "C" void kernel_launch(void* const* d_in, const int* in_sizes, int n_in,
                              void* d_out, int out_size, void* d_ws, size_t ws_size,
                              hipStream_t stream) {
  const int*   edges = (const int*)d_in[0];   // [2,E] flat: row then col
  const float* emb   = (const float*)d_in[1]; // [N,64]
  float*       acc   = (float*)d_out;         // running accumulator lives in d_out

  const int E = in_sizes[0] / 2;
  const int N = in_sizes[1] / DEMB;

  // workspace: deg[N] | dinv2[N] | x[N*64] | buf[N*64]   (~52 MB)
  float* deg   = (float*)d_ws;
  float* dinv2 = deg + N;
  float* x     = dinv2 + N;
  float* buf   = x + (size_t)N * DEMB;

  const int total4 = N * (DEMB / 4);
  const int gTot4  = (total4 + TPB - 1) / TPB;
  const int gE     = (E + TPB - 1) / TPB;
  const int gN     = (N + TPB - 1) / TPB;
  const int gS     = (E + 2 * CHUNK - 1) / (2 * CHUNK);

  lg_init  <<<gTot4, TPB, 0, stream>>>((const float4*)emb, (float4*)acc,
                                       (float4*)x, (float4*)buf, deg, N);
  lg_degree<<<gE,    TPB, 0, stream>>>(edges, deg, E);
  lg_dinv  <<<gN,    TPB, 0, stream>>>(deg, dinv2, N);

  for (int layer = 0; layer < 3; ++layer) {
    lg_scatter<<<gS,    TPB, 0, stream>>>(edges, x, buf, E);
    lg_post   <<<gTot4, TPB, 0, stream>>>((float4*)acc, (float4*)x,
                                          (float4*)buf, dinv2, N, layer == 2);
  }
}